// BiMPM_36610301231109
// MI455X (gfx1250) — compile-verified
//
#include <hip/hip_runtime.h>
#include <math.h>

#define EPS 1e-8f

typedef __attribute__((ext_vector_type(2))) float v2f;
typedef __attribute__((ext_vector_type(8))) float v8f;

// D = A(16x4) * B(4x16) + C, fp32, wave32.
__device__ __forceinline__ v8f wmma4(v2f a, v2f b, v8f c) {
  return __builtin_amdgcn_wmma_f32_16x16x4_f32(false, a, false, b, (short)0, c,
                                               false, false);
}

__device__ __forceinline__ float sigm(float x) { return 1.f / (1.f + expf(-x)); }

// ---------------------------------------------------------------------------
// Elementwise helpers
// ---------------------------------------------------------------------------
__global__ void k_fill(float* __restrict__ p, float v, int n) {
  int i = blockIdx.x * blockDim.x + threadIdx.x;
  if (i < n) p[i] = v;
}

__global__ void k_sqw(const float* __restrict__ W, float* __restrict__ W2, int n) {
  int i = blockIdx.x * blockDim.x + threadIdx.x;
  if (i < n) { float v = W[i]; W2[i] = v * v; }
}

// comb[t, n, d] : n<128 -> premise embed, else hypothesis embed
__global__ void k_embed(const int* __restrict__ prem, const int* __restrict__ hyp,
                        const float* __restrict__ EW, float* __restrict__ comb) {
  int gid = blockIdx.x * blockDim.x + threadIdx.x;
  if (gid >= 64 * 256 * 300) return;
  int t = gid / 76800;
  int rem = gid - t * 76800;
  int n = rem / 300;
  int dd = rem - n * 300;
  int idx = (n < 128) ? prem[t * 128 + n] : hyp[t * 128 + (n - 128)];
  comb[gid] = EW[idx * 300 + dd];
}

// ---------------------------------------------------------------------------
// Fused LSTM step GEMM: G = X @ Wih^T + Hprev @ Whh^T + bias
// One wave per 16x16 tile of G (M x 1200). K = Dx + 300, both multiples of 4.
// ---------------------------------------------------------------------------
__global__ __launch_bounds__(32)
void k_lstm_gemm(const float* __restrict__ X, int Dx,
                 const float* __restrict__ Hprev,
                 const float* __restrict__ Wih, const float* __restrict__ Whh,
                 const float* __restrict__ bias, float* __restrict__ G,
                 int tilesN) {
  int bid = blockIdx.x;
  int tm = bid / tilesN, tn = bid - tm * tilesN;
  int m0 = tm * 16, j0 = tn * 16;
  int lane = threadIdx.x & 31;
  int half = lane >> 4, lm = lane & 15;

  v8f acc = {0.f, 0.f, 0.f, 0.f, 0.f, 0.f, 0.f, 0.f};
  const float* xrow = X + (m0 + lm) * Dx;
  const float* hrow = Hprev + (m0 + lm) * 300;
  const float* wi = Wih + (j0 + lm) * Dx;
  const float* wh = Whh + (j0 + lm) * 300;

  for (int k0 = 0; k0 < Dx; k0 += 4) {
    int ka = k0 + half * 2;
    v2f a, b;
    a.x = xrow[ka];  a.y = xrow[ka + 1];
    b.x = wi[ka];    b.y = wi[ka + 1];
    acc = wmma4(a, b, acc);
  }
  for (int k0 = 0; k0 < 300; k0 += 4) {
    int ka = k0 + half * 2;
    v2f a, b;
    a.x = hrow[ka];  a.y = hrow[ka + 1];
    b.x = wh[ka];    b.y = wh[ka + 1];
    acc = wmma4(a, b, acc);
  }
  int j = j0 + lm;
  float bj = bias[j];
#pragma unroll
  for (int r = 0; r < 8; ++r) {
    int m = m0 + half * 8 + r;
    G[m * 1200 + j] = acc[r] + bj;
  }
}

// gates: i,f,g,o split of G(N x 1200); update c,h; also write hs_out
__global__ void k_lstm_gate(const float* __restrict__ G, float* __restrict__ h,
                            float* __restrict__ c, float* __restrict__ hs_out,
                            int N) {
  int idx = blockIdx.x * blockDim.x + threadIdx.x;
  if (idx >= N * 300) return;
  int n = idx / 300, j = idx - n * 300;
  const float* g = G + n * 1200;
  float iv = sigm(g[j]);
  float fv = sigm(g[300 + j]);
  float gv = tanhf(g[600 + j]);
  float ov = sigm(g[900 + j]);
  float cn = fv * c[idx] + iv * gv;
  float hn = ov * tanhf(cn);
  c[idx] = cn;
  h[idx] = hn;
  hs_out[idx] = hn;
}

// ---------------------------------------------------------------------------
// Perspective-weighted norms:  out[t,b,l] = sqrt(sum_h x[t,b,h]^2 * W2[l,h])
// ---------------------------------------------------------------------------
__global__ void k_wnorm(const float* __restrict__ Xp, int tStride,
                        const float* __restrict__ W2, float* __restrict__ out) {
  int gid = blockIdx.x * blockDim.x + threadIdx.x;
  if (gid >= 64 * 128 * 20) return;
  int t = gid / 2560;
  int rem = gid - t * 2560;
  int b = rem / 20, l = rem - b * 20;
  const float* x = Xp + t * tStride + b * 300;
  const float* w = W2 + l * 300;
  float s = 0.f;
  for (int hh = 0; hh < 300; ++hh) { float v = x[hh]; s += v * v * w[hh]; }
  out[gid] = sqrtf(s);
}

// plain L2 norm over h:  out[t,b] = ||x[t,b,:]||
__global__ void k_vecnorm(const float* __restrict__ Xp, int tStride,
                          float* __restrict__ out) {
  int gid = blockIdx.x * blockDim.x + threadIdx.x;
  if (gid >= 64 * 128) return;
  int t = gid / 128, b = gid - t * 128;
  const float* x = Xp + t * tStride + b * 300;
  float s = 0.f;
  for (int hh = 0; hh < 300; ++hh) { float v = x[hh]; s += v * v; }
  out[gid] = sqrtf(s);
}

// mp_full: m[t,b,colBase+l] = num / (pn*vn + eps), all sums inline
__global__ void k_mp_full(const float* __restrict__ Pp, int tStride,
                          const float* __restrict__ Vf,
                          const float* __restrict__ W2,
                          float* __restrict__ mOut, int colBase) {
  int gid = blockIdx.x * blockDim.x + threadIdx.x;
  if (gid >= 64 * 128 * 20) return;
  int t = gid / 2560;
  int rem = gid - t * 2560;
  int b = rem / 20, l = rem - b * 20;
  const float* p = Pp + t * tStride + b * 300;
  const float* v = Vf + b * 300;
  const float* w = W2 + l * 300;
  float num = 0.f, pn = 0.f, vn = 0.f;
  for (int hh = 0; hh < 300; ++hh) {
    float w2 = w[hh], pv = p[hh], vv = v[hh];
    num += pv * vv * w2;
    pn += pv * pv * w2;
    vn += vv * vv * w2;
  }
  mOut[(t * 128 + b) * 320 + colBase + l] = num / (sqrtf(pn) * sqrtf(vn) + EPS);
}

// mp_pair + max over s. One wave per (b, l, t-tile); WMMA over K=300,
// A[m,k] = P[t0+m,k]*W2[l,k], B[k,n] = H[s0+n,k]; fused max over all s.
__global__ __launch_bounds__(32)
void k_mp_pair_max(const float* __restrict__ Pp, const float* __restrict__ Hm,
                   int tStride, const float* __restrict__ W2,
                   const float* __restrict__ pn, const float* __restrict__ hn,
                   float* __restrict__ mOut, int colBase) {
  int bid = blockIdx.x;
  int tt = bid & 3;
  int l = (bid >> 2) % 20;
  int b = bid / 80;
  int t0 = tt * 16;
  int lane = threadIdx.x & 31;
  int half = lane >> 4, lm = lane & 15;

  const float* w = W2 + l * 300;
  const float* prow = Pp + (t0 + lm) * tStride + b * 300;
  float pnr[8];
#pragma unroll
  for (int r = 0; r < 8; ++r)
    pnr[r] = pn[((t0 + half * 8 + r) * 128 + b) * 20 + l];
  float mx[8];
#pragma unroll
  for (int r = 0; r < 8; ++r) mx[r] = -3.402823466e38f;

  for (int s0 = 0; s0 < 64; s0 += 16) {
    const float* hrow = Hm + (s0 + lm) * tStride + b * 300;
    float hns = hn[((s0 + lm) * 128 + b) * 20 + l];
    v8f acc = {0.f, 0.f, 0.f, 0.f, 0.f, 0.f, 0.f, 0.f};
    for (int k0 = 0; k0 < 300; k0 += 4) {
      int ka = k0 + half * 2;
      v2f a, bv;
      a.x = prow[ka] * w[ka];
      a.y = prow[ka + 1] * w[ka + 1];
      bv.x = hrow[ka];
      bv.y = hrow[ka + 1];
      acc = wmma4(a, bv, acc);
    }
#pragma unroll
    for (int r = 0; r < 8; ++r) {
      float val = acc[r] / (pnr[r] * hns + EPS);
      mx[r] = fmaxf(mx[r], val);
    }
  }
  for (int off = 8; off; off >>= 1) {
#pragma unroll
    for (int r = 0; r < 8; ++r)
      mx[r] = fmaxf(mx[r], __shfl_xor(mx[r], off, 16));
  }
  if (lm == 0) {
#pragma unroll
    for (int r = 0; r < 8; ++r) {
      int t = t0 + half * 8 + r;
      mOut[(t * 128 + b) * 320 + colBase + l] = mx[r];
    }
  }
}

// cosine attention: att[b][t][s], row sums, row argmax. One wave per (b,t-tile)
__global__ __launch_bounds__(32)
void k_att(const float* __restrict__ Pp, const float* __restrict__ Hm,
           int tStride, const float* __restrict__ pnorm,
           const float* __restrict__ hnorm, float* __restrict__ attbuf,
           float* __restrict__ attsum, int* __restrict__ attidx) {
  int bid = blockIdx.x;
  int tt = bid & 3;
  int b = bid >> 2;
  int t0 = tt * 16;
  int lane = threadIdx.x & 31;
  int half = lane >> 4, lm = lane & 15;

  const float* prow = Pp + (t0 + lm) * tStride + b * 300;
  float invp[8];
#pragma unroll
  for (int r = 0; r < 8; ++r)
    invp[r] = 1.f / (pnorm[(t0 + half * 8 + r) * 128 + b] + EPS);
  float rowsum[8], best[8];
  int bidx[8];
#pragma unroll
  for (int r = 0; r < 8; ++r) { rowsum[r] = 0.f; best[r] = -3.402823466e38f; bidx[r] = 0; }

  for (int s0 = 0; s0 < 64; s0 += 16) {
    const float* hrow = Hm + (s0 + lm) * tStride + b * 300;
    float invh = 1.f / (hnorm[(s0 + lm) * 128 + b] + EPS);
    v8f acc = {0.f, 0.f, 0.f, 0.f, 0.f, 0.f, 0.f, 0.f};
    for (int k0 = 0; k0 < 300; k0 += 4) {
      int ka = k0 + half * 2;
      v2f a, bv;
      a.x = prow[ka];      a.y = prow[ka + 1];
      bv.x = hrow[ka];     bv.y = hrow[ka + 1];
      acc = wmma4(a, bv, acc);
    }
#pragma unroll
    for (int r = 0; r < 8; ++r) {
      int t = t0 + half * 8 + r;
      int s = s0 + lm;
      float a = acc[r] * invp[r] * invh;
      attbuf[(b * 64 + t) * 64 + s] = a;
      rowsum[r] += a;
      if (a > best[r]) { best[r] = a; bidx[r] = s; }
    }
  }
  for (int off = 8; off; off >>= 1) {
#pragma unroll
    for (int r = 0; r < 8; ++r) {
      rowsum[r] += __shfl_xor(rowsum[r], off, 16);
      float bv = __shfl_xor(best[r], off, 16);
      int bi = __shfl_xor(bidx[r], off, 16);
      if (bv > best[r] || (bv == best[r] && bi < bidx[r])) {
        best[r] = bv;
        bidx[r] = bi;
      }
    }
  }
  if (lm == 0) {
#pragma unroll
    for (int r = 0; r < 8; ++r) {
      int t = t0 + half * 8 + r;
      attsum[t * 128 + b] = rowsum[r];
      attidx[t * 128 + b] = bidx[r];
    }
  }
}

// hmean[t,b,:] = sum_s (att[t,s]/(rowsum+eps)) * H[s,b,:]   (WMMA, K=64)
__global__ __launch_bounds__(32)
void k_hmean(const float* __restrict__ attbuf, const float* __restrict__ attsum,
             const float* __restrict__ Hm, int tStride,
             float* __restrict__ hmean) {
  int bid = blockIdx.x;
  int ht = bid % 19;
  int tt = (bid / 19) & 3;
  int b = bid / 76;
  int t0 = tt * 16, h0 = ht * 16;
  int lane = threadIdx.x & 31;
  int half = lane >> 4, lm = lane & 15;

  float scale = 1.f / (attsum[(t0 + lm) * 128 + b] + EPS);
  const float* arow = attbuf + (b * 64 + t0 + lm) * 64;
  int hc = h0 + lm;
  int hcl = hc < 300 ? hc : 299;  // clamp loads; store is guarded

  v8f acc = {0.f, 0.f, 0.f, 0.f, 0.f, 0.f, 0.f, 0.f};
  for (int k0 = 0; k0 < 64; k0 += 4) {
    int ka = k0 + half * 2;
    v2f a, bv;
    a.x = arow[ka] * scale;
    a.y = arow[ka + 1] * scale;
    bv.x = Hm[ka * tStride + b * 300 + hcl];
    bv.y = Hm[(ka + 1) * tStride + b * 300 + hcl];
    acc = wmma4(a, bv, acc);
  }
  if (hc < 300) {
#pragma unroll
    for (int r = 0; r < 8; ++r) {
      int t = t0 + half * 8 + r;
      hmean[t * 38400 + b * 300 + hc] = acc[r];
    }
  }
}

// mp_elem: q comes either from Q[t,b,:] or gathered H[idx[t,b],b,:]
__global__ void k_mp_elem(const float* __restrict__ Pp, int pStride,
                          const float* __restrict__ Q, int qStride,
                          const int* __restrict__ idx,
                          const float* __restrict__ Hm, int hStride,
                          const float* __restrict__ W2,
                          float* __restrict__ mOut, int colBase) {
  int gid = blockIdx.x * blockDim.x + threadIdx.x;
  if (gid >= 64 * 128 * 20) return;
  int t = gid / 2560;
  int rem = gid - t * 2560;
  int b = rem / 20, l = rem - b * 20;
  const float* p = Pp + t * pStride + b * 300;
  const float* q = idx ? (Hm + idx[t * 128 + b] * hStride + b * 300)
                       : (Q + t * qStride + b * 300);
  const float* w = W2 + l * 300;
  float num = 0.f, pn = 0.f, qn = 0.f;
  for (int hh = 0; hh < 300; ++hh) {
    float w2 = w[hh], pv = p[hh], qv = q[hh];
    num += pv * qv * w2;
    pn += pv * pv * w2;
    qn += qv * qv * w2;
  }
  mOut[(t * 128 + b) * 320 + colBase + l] = num / (sqrtf(pn) * sqrtf(qn) + EPS);
}

// ---------------------------------------------------------------------------
// MLP head: bn -> W1 -> relu -> bn -> W2 -> relu -> bn -> out. One block/row.
// ---------------------------------------------------------------------------
__global__ __launch_bounds__(256)
void k_mlp(const float* __restrict__ af, const float* __restrict__ ab,
           const float* __restrict__ bng, const float* __restrict__ bnb,
           const float* __restrict__ W1, const float* __restrict__ b1,
           const float* __restrict__ W2, const float* __restrict__ b2,
           const float* __restrict__ oW, const float* __restrict__ ob,
           float* __restrict__ out) {
  __shared__ float xs[600];
  __shared__ float ys[600];
  int b = blockIdx.x;
  int tid = threadIdx.x;
  const float inv = 0.99999500003749f;  // 1/sqrt(1+1e-5)

  for (int j = tid; j < 600; j += 256) {
    float v = (j < 300) ? af[b * 300 + j] : ab[b * 300 + (j - 300)];
    xs[j] = bng[j] * v * inv + bnb[j];
  }
  __syncthreads();
  for (int j = tid; j < 600; j += 256) {
    float s = b1[j];
    const float* wr = W1 + j * 600;
    for (int k = 0; k < 600; ++k) s += xs[k] * wr[k];
    ys[j] = fmaxf(s, 0.f);
  }
  __syncthreads();
  for (int j = tid; j < 600; j += 256)
    xs[j] = bng[600 + j] * ys[j] * inv + bnb[600 + j];
  __syncthreads();
  for (int j = tid; j < 600; j += 256) {
    float s = b2[j];
    const float* wr = W2 + j * 600;
    for (int k = 0; k < 600; ++k) s += xs[k] * wr[k];
    ys[j] = fmaxf(s, 0.f);
  }
  __syncthreads();
  for (int j = tid; j < 600; j += 256)
    xs[j] = bng[1200 + j] * ys[j] * inv + bnb[1200 + j];
  __syncthreads();
  if (tid < 3) {
    float s = ob[tid];
    const float* wr = oW + tid * 600;
    for (int k = 0; k < 600; ++k) s += xs[k] * wr[k];
    out[b * 3 + tid] = s;
  }
}

// ---------------------------------------------------------------------------
// Orchestration
// ---------------------------------------------------------------------------
extern "C" void kernel_launch(void* const* d_in, const int* in_sizes, int n_in,
                              void* d_out, int out_size, void* d_ws,
                              size_t ws_size, hipStream_t stream) {
  const int* prem = (const int*)d_in[0];
  const int* hyp = (const int*)d_in[1];
  const float* EW = (const float*)d_in[2];
  const float* cWih = (const float*)d_in[3];
  const float* cWhh = (const float*)d_in[4];
  const float* cb = (const float*)d_in[5];
  const float* mpW = (const float*)d_in[6];
  const float* aWih_f = (const float*)d_in[7];
  const float* aWhh_f = (const float*)d_in[8];
  const float* ab_f = (const float*)d_in[9];
  const float* aWih_b = (const float*)d_in[10];
  const float* aWhh_b = (const float*)d_in[11];
  const float* ab_b = (const float*)d_in[12];
  const float* bng = (const float*)d_in[13];
  const float* bnb = (const float*)d_in[14];
  const float* W1 = (const float*)d_in[15];
  const float* b1 = (const float*)d_in[16];
  const float* W2mlp = (const float*)d_in[17];
  const float* b2 = (const float*)d_in[18];
  const float* oW = (const float*)d_in[19];
  const float* ob = (const float*)d_in[20];
  float* outp = (float*)d_out;

  // workspace layout (floats), ~85.5 MB total
  float* ws = (float*)d_ws;
  float* comb = ws;                        // 64*256*300 = 4,915,200
  float* hs_fw = comb + 4915200;           // 4,915,200
  float* hs_bw = hs_fw + 4915200;          // 4,915,200
  float* h_cur = hs_bw + 4915200;          // 76,800
  float* c_cur = h_cur + 76800;            // 76,800
  float* g_buf = c_cur + 76800;            // 307,200
  float* W2all = g_buf + 307200;           // 8*20*300 = 48,000
  float* m_buf = W2all + 48000;            // 64*128*320 = 2,621,440
  float* pn_buf = m_buf + 2621440;         // 163,840
  float* hn_buf = pn_buf + 163840;         // 163,840
  float* pnorm_b = hn_buf + 163840;        // 8,192
  float* hnorm_b = pnorm_b + 8192;         // 8,192
  float* attbuf = hnorm_b + 8192;          // 128*64*64 = 524,288
  float* attsum = attbuf + 524288;         // 8,192
  int* attidx = (int*)(attsum + 8192);     // 8,192 ints
  float* hmean = (float*)(attidx + 8192);  // 64*128*300 = 2,457,600
  float* agg_h = hmean + 2457600;          // 38,400
  float* agg_c = agg_h + 38400;            // 38,400
  float* agg_hb = agg_c + 38400;           // 38,400
  float* agg_cb = agg_hb + 38400;          // 38,400

  // 1) embed + squared perspective weights
  k_embed<<<19200, 256, 0, stream>>>(prem, hyp, EW, comb);
  k_sqw<<<(48000 + 255) / 256, 256, 0, stream>>>(mpW, W2all, 48000);

  // 2) encoder LSTM forward (batch 256): fused gemm per step + gate update
  k_fill<<<600, 256, 0, stream>>>(h_cur, 0.f, 153600);  // h_cur+c_cur
  for (int s = 0; s < 64; ++s) {
    k_lstm_gemm<<<16 * 75, 32, 0, stream>>>(comb + s * 76800, 300, h_cur, cWih,
                                            cWhh, cb, g_buf, 75);
    k_lstm_gate<<<300, 256, 0, stream>>>(g_buf, h_cur, c_cur,
                                         hs_fw + s * 76800, 256);
  }
  // encoder LSTM backward (same cell, reversed time)
  k_fill<<<600, 256, 0, stream>>>(h_cur, 0.f, 153600);
  for (int s = 0; s < 64; ++s) {
    int t = 63 - s;
    k_lstm_gemm<<<16 * 75, 32, 0, stream>>>(comb + t * 76800, 300, h_cur, cWih,
                                            cWhh, cb, g_buf, 75);
    k_lstm_gate<<<300, 256, 0, stream>>>(g_buf, h_cur, c_cur,
                                         hs_bw + t * 76800, 256);
  }

  const float* p_fw = hs_fw;
  const float* h_fw = hs_fw + 38400;
  const float* p_bw = hs_bw;
  const float* h_bw = hs_bw + 38400;

  // 3) multi-perspective matching (4 directions)
  auto match_dir = [&](const float* Pp, const float* Hm, const float* Vf,
                       int wbase, int colBase) {
    const float* W2f = W2all + (wbase + 0) * 6000;
    const float* W2m = W2all + (wbase + 1) * 6000;
    const float* W2a = W2all + (wbase + 2) * 6000;
    const float* W2ma = W2all + (wbase + 3) * 6000;
    k_mp_full<<<640, 256, 0, stream>>>(Pp, 76800, Vf, W2f, m_buf, colBase);
    k_wnorm<<<640, 256, 0, stream>>>(Pp, 76800, W2m, pn_buf);
    k_wnorm<<<640, 256, 0, stream>>>(Hm, 76800, W2m, hn_buf);
    k_mp_pair_max<<<128 * 20 * 4, 32, 0, stream>>>(Pp, Hm, 76800, W2m, pn_buf,
                                                   hn_buf, m_buf, colBase + 20);
    k_vecnorm<<<32, 256, 0, stream>>>(Pp, 76800, pnorm_b);
    k_vecnorm<<<32, 256, 0, stream>>>(Hm, 76800, hnorm_b);
    k_att<<<512, 32, 0, stream>>>(Pp, Hm, 76800, pnorm_b, hnorm_b, attbuf,
                                  attsum, attidx);
    k_hmean<<<128 * 4 * 19, 32, 0, stream>>>(attbuf, attsum, Hm, 76800, hmean);
    k_mp_elem<<<640, 256, 0, stream>>>(Pp, 76800, hmean, 38400, nullptr,
                                       nullptr, 0, W2a, m_buf, colBase + 40);
    k_mp_elem<<<640, 256, 0, stream>>>(Pp, 76800, nullptr, 0, attidx, Hm, 76800,
                                       W2ma, m_buf, colBase + 60);
  };

  match_dir(p_fw, h_fw, hs_fw + 63 * 76800 + 38400, 0, 0);     // vs h_fw[-1]
  match_dir(p_bw, h_bw, hs_bw + 38400, 4, 80);                 // vs h_bw[0]
  match_dir(h_fw, p_fw, hs_fw + 63 * 76800, 0, 160);           // vs p_fw[-1]
  match_dir(h_bw, p_bw, hs_bw, 4, 240);                        // vs p_bw[0]

  // 4) aggregation LSTMs (batch 128, in=320), keep final hidden only
  k_fill<<<300, 256, 0, stream>>>(agg_h, 0.f, 76800);  // agg_h + agg_c
  for (int s = 0; s < 64; ++s) {
    k_lstm_gemm<<<8 * 75, 32, 0, stream>>>(m_buf + s * 40960, 320, agg_h,
                                           aWih_f, aWhh_f, ab_f, g_buf, 75);
    k_lstm_gate<<<150, 256, 0, stream>>>(g_buf, agg_h, agg_c, agg_h, 128);
  }
  k_fill<<<300, 256, 0, stream>>>(agg_hb, 0.f, 76800);  // agg_hb + agg_cb
  for (int s = 0; s < 64; ++s) {
    int t = 63 - s;
    k_lstm_gemm<<<8 * 75, 32, 0, stream>>>(m_buf + t * 40960, 320, agg_hb,
                                           aWih_b, aWhh_b, ab_b, g_buf, 75);
    k_lstm_gate<<<150, 256, 0, stream>>>(g_buf, agg_hb, agg_cb, agg_hb, 128);
  }

  // 5) MLP head -> (128, 3)
  k_mlp<<<128, 256, 0, stream>>>(agg_h, agg_hb, bng, bnb, W1, b1, W2mlp, b2, oW,
                                 ob, outp);
}